// SimplestSpline_30124900614364
// MI455X (gfx1250) — compile-verified
//
#include <hip/hip_runtime.h>

// SimplestSpline: per-pixel piecewise-linear curve, pure streaming op.
// ~201 MB HBM traffic, ~0.13 GFLOP -> HBM-bound (floor ~8.6 us @ 23.3 TB/s).
// Fast path (exact tiling, reference shape): b128 non-temporal loads/stores,
// per-(b,c) LDS LUT of (slope, intercept) float2 pairs -> one ds_load_b64 +
// one v_fma + one u32 cmp/cndmask per element. Generic shapes fall back to a
// compact scalar grid-stride kernel (never used for the reference shape).

typedef float v4f __attribute__((ext_vector_type(4)));
typedef float v2f __attribute__((ext_vector_type(2)));

#define NK    16            // knots from params per channel
#define NSEG  17            // linear segments (knots + 1)
#define BLOCK 256           // 8 waves on wave32
#define PER_THREAD 8        // float4s per thread -> 2048 float4 per block tile

// Build the per-(b,c) {slope, intercept} LUT in LDS. val = c0 + x*s.
__device__ __forceinline__ void build_lut(v2f* lut, const float* __restrict__ params,
                                          int p, int tid) {
    const int b = p / 3;
    const int c = p - 3 * b;
    const float* __restrict__ pp = params + b * (3 * NK) + c * NK;
    if (tid < NSEG) {
        // ys_full = [0, params[0..15], 1]; evenly spaced knots, dx = 1/17
        float ylo = (tid == 0)  ? 0.0f : pp[tid - 1];
        float yhi = (tid == NK) ? 1.0f : pp[tid];
        const float dx = 1.0f / (float)NSEG;
        float s  = (yhi - ylo) / dx;                     // slope of segment
        float c0 = yhi - (float)(tid + 1) * dx * s;      // val = c0 + x*s
        v2f e; e.x = s; e.y = c0;
        lut[tid] = e;
    }
    __syncthreads();
}

__device__ __forceinline__ float eval1(const v2f* lut, float xv) {
    int seg = (int)(xv * (float)NSEG);                   // trunc == floor, x>=0
    seg = seg < 0 ? 0 : (seg > NSEG - 1 ? NSEG - 1 : seg);  // v_med3_i32
    v2f sc = lut[seg];                                   // one ds_load_b64
    float val = __builtin_fmaf(xv, sc.x, sc.y);          // one v_fma
    // covered = (x >= 0) && (x < 1)  <=>  bits(x) < 0x3f800000 (u32)
    unsigned bits = __builtin_bit_cast(unsigned, xv);
    return (bits < 0x3f800000u) ? val : 0.0f;            // 1 cmp + 1 cndmask
}

// Fast path: plane is an exact multiple of BLOCK*PER_THREAD float4s.
__global__ __launch_bounds__(BLOCK)
void spline_fast_kernel(const float* __restrict__ raw,
                        const float* __restrict__ params,
                        float* __restrict__ out,
                        int planeVec)   // float4 count per (b,c) plane
{
    // 17 float2 entries span 34 distinct LDS banks -> divergent ds_load_b64
    // gather is conflict-free (same-address reads broadcast).
    __shared__ v2f lut[NSEG];
    const int p = blockIdx.y;
    const int tid = threadIdx.x;
    build_lut(lut, params, p, tid);

    const size_t planeOff = (size_t)p * (size_t)planeVec;
    const v4f* __restrict__ in4  = (const v4f*)raw + planeOff;
    v4f*       __restrict__ out4 = (v4f*)out       + planeOff;

    const int base = blockIdx.x * (BLOCK * PER_THREAD) + tid;

    // Phase 1: 8 independent b128 NT loads back-to-back (max MLP).
    v4f v[PER_THREAD];
#pragma unroll
    for (int k = 0; k < PER_THREAD; ++k)
        v[k] = __builtin_nontemporal_load(&in4[base + k * BLOCK]);

    // Phase 2: LUT-evaluate, stream out with NT stores.
#pragma unroll
    for (int k = 0; k < PER_THREAD; ++k) {
        v4f x = v[k];
        v4f r;
#pragma unroll
        for (int e = 0; e < 4; ++e)
            r[e] = eval1(lut, x[e]);
        __builtin_nontemporal_store(r, &out4[base + k * BLOCK]);
    }
}

// Generic fallback: scalar grid-stride over one plane per blockIdx.y.
__global__ __launch_bounds__(BLOCK)
void spline_tail_kernel(const float* __restrict__ raw,
                        const float* __restrict__ params,
                        float* __restrict__ out,
                        int plane)      // elements per (b,c) plane
{
    __shared__ v2f lut[NSEG];
    const int p = blockIdx.y;
    const int tid = threadIdx.x;
    build_lut(lut, params, p, tid);

    const size_t planeOff = (size_t)p * (size_t)plane;
    const float* __restrict__ in  = raw + planeOff;
    float*       __restrict__ o   = out + planeOff;

    const int stride = gridDim.x * BLOCK;
    for (int i = blockIdx.x * BLOCK + tid; i < plane; i += stride)
        o[i] = eval1(lut, in[i]);
}

extern "C" void kernel_launch(void* const* d_in, const int* in_sizes, int n_in,
                              void* d_out, int out_size, void* d_ws, size_t ws_size,
                              hipStream_t stream) {
    (void)d_ws; (void)ws_size; (void)n_in; (void)out_size;

    const float* raw    = (const float*)d_in[0];
    const float* params = (const float*)d_in[1];
    float*       out    = (float*)d_out;

    const int n        = in_sizes[0];          // total raw elements
    const int npar     = in_sizes[1];          // B * 3 * NK
    const int B        = npar / (3 * NK);      // 8
    const int planes   = B * 3;                // 24
    const int plane    = n / planes;           // 1024*1024
    const int planeVec = plane / 4;            // float4s per plane

    const int chunk = BLOCK * PER_THREAD;      // 2048 float4 per block tile

    if ((plane % 4 == 0) && (planeVec % chunk == 0)) {
        // Reference shape: 262144 float4 / plane = 128 tiles, 24 planes.
        dim3 grid(planeVec / chunk, planes);
        spline_fast_kernel<<<grid, BLOCK, 0, stream>>>(raw, params, out, planeVec);
    } else {
        int bx = (plane + BLOCK - 1) / BLOCK;
        if (bx > 512) bx = 512;
        dim3 grid(bx, planes);
        spline_tail_kernel<<<grid, BLOCK, 0, stream>>>(raw, params, out, plane);
    }
}